// CTCLoss_30236569763915
// MI455X (gfx1250) — compile-verified
//
#include <hip/hip_runtime.h>
#include <hip/hip_bf16.h>
#include <math.h>

#define NEG (-1e30f)
#define WAVES_PER_BLOCK 8

typedef unsigned int u32x4 __attribute__((ext_vector_type(4)));
typedef int          i32x8 __attribute__((ext_vector_type(8)));
typedef int          i32x4 __attribute__((ext_vector_type(4)));
typedef float        f32x2 __attribute__((ext_vector_type(2)));
typedef float        f32x8 __attribute__((ext_vector_type(8)));

// ---------------------------------------------------------------------------
// Phase 1: per-(t,b) log-sum-exp over V.  One wave per (t,b) row.
// Row is DMA'd into LDS by the Tensor Data Mover; max via shuffle butterfly;
// the 32-lane sum all-reduce is done with one V_WMMA_F32_16X16X4_F32
// (B = ones => D[m][n] = s_m + s_{m+16}) + 8 component adds + 1 shuffle.
// Writes lognorm[t*B + b] = max + log(sum exp(x - max)).
// ---------------------------------------------------------------------------
__global__ __launch_bounds__(WAVES_PER_BLOCK * 32)
void ctc_lse_kernel(const float* __restrict__ acts,
                    float* __restrict__ lognorm,
                    int TB, int V) {
    __shared__ float smem[WAVES_PER_BLOCK * 2048];  // 8 KB per wave
    const int wave = threadIdx.x >> 5;
    const int lane = threadIdx.x & 31;
    const int p = blockIdx.x * WAVES_PER_BLOCK + wave;   // p = t*B + b
    if (p >= TB) return;                                  // whole-wave uniform exit

    // ---- Tensor DMA: global row -> LDS (CDNA5 TDM, D# per ISA 08 §8) ----
    unsigned lds_off = (unsigned)(size_t)(&smem[wave * 2048]); // low 32b = LDS offset
    unsigned long long ga = (unsigned long long)(const void*)(acts + (size_t)p * V);

    u32x4 g0;
    g0[0] = 1u;                                          // count=1 (valid user D#)
    g0[1] = lds_off;                                     // lds_addr [63:32]
    g0[2] = (unsigned)(ga & 0xFFFFFFFFull);              // global_addr lo
    g0[3] = (unsigned)((ga >> 32) & 0x1FFFFFFull)        // global_addr [56:32]
          | (2u << 30);                                  // type=2 ("image")

    i32x8 g1;
    g1[0] = 0x00020000;                                  // data_size=2 -> 4 bytes
    g1[1] = (int)(((unsigned)V & 0xFFFFu) << 16);        // tensor_dim0 lo16 @ bits 63:48
    g1[2] = (int)(1u << 16);                             // tensor_dim1=1 @ bits 95:80
    g1[3] = (int)(((unsigned)V & 0xFFFFu) << 16);        // tile_dim0=V @ bits 127:112
    g1[4] = 1;                                           // tile_dim1=1 @ bits 143:128
    g1[5] = V;                                           // tensor_dim0_stride lo32
    g1[6] = 0;
    g1[7] = 0;

    i32x4 gz4 = {0, 0, 0, 0};                            // groups 2/3 unused (2D)
    i32x8 gz8 = {0, 0, 0, 0, 0, 0, 0, 0};
    __builtin_amdgcn_tensor_load_to_lds(g0, g1, gz4, gz4, gz8, 0);
    __builtin_amdgcn_s_wait_tensorcnt(0);

    // ---- pass 1: max (shuffle butterfly) ----
    const float4* row = (const float4*)(&smem[wave * 2048]);
    const int n4 = V >> 2;
    float m = -INFINITY;
    for (int i = lane; i < n4; i += 32) {
        float4 x = row[i];
        m = fmaxf(m, fmaxf(fmaxf(x.x, x.y), fmaxf(x.z, x.w)));
    }
    #pragma unroll
    for (int off = 16; off > 0; off >>= 1)
        m = fmaxf(m, __shfl_xor(m, off));

    // ---- pass 2: sum exp(x - m) ----
    float s = 0.f;
    for (int i = lane; i < n4; i += 32) {
        float4 x = row[i];
        s += expf(x.x - m) + expf(x.y - m) + expf(x.z - m) + expf(x.w - m);
    }

    // ---- 32-lane sum all-reduce via WMMA f32 16x16x4 ----
    // A[m][0] = s_m (lanes 0-15 VGPR0), A[m][2] = s_{m+16} (lanes 16-31 VGPR0);
    // A[m][1] = A[m][3] = 0.  B = ones.  D[m][n] = s_m + s_{m+16}.
    f32x2 amat; amat[0] = s;    amat[1] = 0.0f;
    f32x2 bone; bone[0] = 1.0f; bone[1] = 1.0f;
    f32x8 cacc = {0.f, 0.f, 0.f, 0.f, 0.f, 0.f, 0.f, 0.f};
    f32x8 d = __builtin_amdgcn_wmma_f32_16x16x4_f32(
        false, amat, false, bone, (short)0, cacc, false, false);
    // lane n (0-15) holds rows {0..7}: sum_r d[r] = sum_{m=0..7}(s_m + s_{m+16});
    // lane n (16-31) holds rows {8..15}: sum = sum_{m=8..15}(s_m + s_{m+16}).
    float partial = ((d[0] + d[1]) + (d[2] + d[3])) + ((d[4] + d[5]) + (d[6] + d[7]));
    float total = partial + __shfl_xor(partial, 16);

    if (lane == 0) lognorm[p] = m + logf(total);
}

// ---------------------------------------------------------------------------
// Phase 2: alpha recursion.  One block per batch element b; thread = state s.
// lp(t,s) gathered from acts and prefetched 4 timesteps ahead (register ring)
// to hide global latency behind the sequential exp/log chain.
// ---------------------------------------------------------------------------
__global__ void ctc_alpha_kernel(const float* __restrict__ acts,
                                 const int* __restrict__ labels,
                                 const int* __restrict__ act_lens,
                                 const int* __restrict__ label_lens,
                                 const float* __restrict__ lognorm,
                                 float* __restrict__ loss_b,
                                 int T, int B, int V, int L) {
    const int b = blockIdx.x;
    const int S = 2 * L + 1;
    const int s = threadIdx.x;
    const bool active = (s < S);

    __shared__ float buf[2][1026];

    // extended label + allow-2 transition flag
    int ext = 0;
    bool allow2 = false;
    if (active && (s & 1)) {
        int li = s >> 1;                 // (s-1)/2 for odd s
        ext = labels[b * L + li];        // non-blank label (>=1)
        if (s >= 3) allow2 = (ext != labels[b * L + li - 1]);
    }
    const int alen = act_lens[b];
    const size_t strideT = (size_t)B * V;
    const size_t base_bv = (size_t)b * V + (size_t)ext;

    // prefetch ring, depth 4
    float a_pf[4], n_pf[4];
    #pragma unroll
    for (int d = 0; d < 4; ++d) {
        if (d < T) {
            if (active) a_pf[d] = acts[(size_t)d * strideT + base_bv];
            n_pf[d] = lognorm[(size_t)d * B + b];
        }
    }

    // t = 0: alpha[0] = lp(0,0), alpha[1] = lp(0,1), rest NEG
    float a_init = NEG;
    if (s < 2) a_init = a_pf[0] - n_pf[0];
    if (active) buf[0][s] = a_init;
    __syncthreads();
    if (4 < T) {                         // refill slot 0 (consumed by init)
        if (active) a_pf[0] = acts[(size_t)4 * strideT + base_bv];
        n_pf[0] = lognorm[(size_t)4 * B + b];
    }

    int cur = 0;
    for (int t = 1; t < T; ++t) {
        const int slot = t & 3;
        const float lp = a_pf[slot] - n_pf[slot];
        if (active) {
            float a  = buf[cur][s];
            float a1 = (s >= 1) ? buf[cur][s - 1] : NEG;
            float a2 = allow2   ? buf[cur][s - 2] : NEG;
            float mm = fmaxf(a, fmaxf(a1, a2));
            float r = mm + logf(expf(a - mm) + expf(a1 - mm) + expf(a2 - mm)) + lp;
            if (t >= alen) r = a;        // freeze ended sequences
            buf[cur ^ 1][s] = r;
        }
        __syncthreads();
        cur ^= 1;
        const int tp = t + 4;            // prefetch next ring slot
        if (tp < T) {
            if (active) a_pf[slot] = acts[(size_t)tp * strideT + base_bv];
            n_pf[slot] = lognorm[(size_t)tp * B + b];
        }
    }

    if (threadIdx.x == 0) {
        int slast = 2 * label_lens[b];
        float v1 = buf[cur][slast];
        float v2 = buf[cur][slast - 1];
        float mm = fmaxf(v1, v2);
        loss_b[b] = -(mm + logf(expf(v1 - mm) + expf(v2 - mm)));
    }
}

// ---------------------------------------------------------------------------
// Phase 3: deterministic fixed-order sum of per-batch losses.
// ---------------------------------------------------------------------------
__global__ void ctc_sum_kernel(const float* __restrict__ loss_b,
                               float* __restrict__ out, int B) {
    if (threadIdx.x == 0 && blockIdx.x == 0) {
        float acc = 0.f;
        for (int i = 0; i < B; ++i) acc += loss_b[i];
        out[0] = acc;
    }
}

// ---------------------------------------------------------------------------
extern "C" void kernel_launch(void* const* d_in, const int* in_sizes, int n_in,
                              void* d_out, int out_size, void* d_ws, size_t ws_size,
                              hipStream_t stream) {
    const float* acts       = (const float*)d_in[0];
    const int*   labels     = (const int*)d_in[1];
    const int*   act_lens   = (const int*)d_in[2];
    const int*   label_lens = (const int*)d_in[3];

    const int B = in_sizes[2];                 // act_lens has one entry per example
    const int L = in_sizes[1] / B;             // uniform label length
    const int V = 2048;                        // vocabulary size (fixed by reference)
    const int T = (int)((long long)in_sizes[0] / ((long long)B * V));
    const int TB = T * B;

    float* lognorm = (float*)d_ws;             // T*B floats
    float* loss_b  = lognorm + (size_t)TB;     // B floats

    // Phase 1: log-sum-exp per (t,b), one wave each, TDM-staged rows.
    const int blocks1 = (TB + WAVES_PER_BLOCK - 1) / WAVES_PER_BLOCK;
    ctc_lse_kernel<<<blocks1, WAVES_PER_BLOCK * 32, 0, stream>>>(acts, lognorm, TB, V);

    // Phase 2: alpha recursion, one block per example.
    const int S = 2 * L + 1;
    int bdim = ((S + 31) / 32) * 32;
    if (bdim > 1024) bdim = 1024;
    ctc_alpha_kernel<<<B, bdim, 0, stream>>>(acts, labels, act_lens, label_lens,
                                             lognorm, loss_b, T, B, V, L);

    // Phase 3: deterministic final reduction.
    ctc_sum_kernel<<<1, 32, 0, stream>>>(loss_b, (float*)d_out, B);
}